// OutputLayer_3100966388047
// MI455X (gfx1250) — compile-verified
//
#include <hip/hip_runtime.h>

// LIF output layer: out[b,f,n] = step( reset( v0[b,f,n]*DECAY + spikes[b,f] ) )
// Purely elementwise + broadcast: 268 MB of HBM traffic, ~0.17 GFLOP.
// => HBM-bound (~11.5 us floor @ 23.3 TB/s). Optimal CDNA5 shape:
//   - double-buffered async global->LDS staging (ASYNCcnt pipeline)
//   - 128-bit per-lane accesses everywhere
//   - NT temporal hints (streaming set 256 MB > 192 MB L2)
//   - wave32 blocks of 256 threads (8 waves)

typedef __attribute__((ext_vector_type(4))) float v4f;

namespace {
constexpr int      kB = 32;
constexpr int      kF = 4096;
constexpr int      kN = 256;
constexpr unsigned kTotal = (unsigned)kB * kF * kN;          // 33,554,432 floats
constexpr unsigned kNV4   = kTotal / 4;                      // 8,388,608 float4
constexpr int      kBlock = 256;                             // 8 waves (wave32)
constexpr int      kTilesPerBlock = 8;                       // 8 x 4KB tiles per block
constexpr unsigned kGrid  = kNV4 / ((unsigned)kBlock * kTilesPerBlock); // 4096
constexpr float    kDecay  = 0.5f;
constexpr float    kThresh = 0.5f;

static_assert(kNV4 % ((unsigned)kBlock * kTilesPerBlock) == 0, "exact tiling");
static_assert(kN % 4 == 0, "float4 along N");
}  // namespace

__global__ __launch_bounds__(kBlock) void lif_stream_kernel(
    const float* __restrict__ spikes,   // [B, F]
    const float* __restrict__ v0,       // [B, F, N]
    float* __restrict__ out)            // [B, F, N]
{
  // Double-buffered staging area: 2 x 256 lanes x 16B = 8 KB LDS.
  __shared__ v4f buf[2][kBlock];

  const unsigned tid      = threadIdx.x;
  const unsigned tileBase = (unsigned)blockIdx.x * (unsigned)kTilesPerBlock;

  // gfx1250 flat-address aperture: for LDS pointers, addr[31:0] IS the
  // wave-relative LDS byte offset (addr[63:32] == SHARED_BASE). Deriving the
  // async-DMA destination from the pointer (ptrtoint) also CAPTURES `buf`,
  // so the "memory"-clobbering asms below are assumed to write it and the
  // compiler cannot fold the subsequent ds_load_b128 reads to undef
  // (which round 1 did: ds=0, QNaN constants).
  unsigned lds_base = (unsigned)(unsigned long long)(const void*)&buf[0][0];
  asm volatile("" : "+v"(lds_base) :: "memory");   // escape + opaque

  // ---- prime the async pipeline: tile 0 -> LDS buffer 0 ----
  {
    const unsigned g       = tileBase * (unsigned)kBlock + tid;      // float4 index
    const unsigned lds_off = lds_base + tid * (unsigned)sizeof(v4f); // &buf[0][tid]
    const unsigned long long src =
        (unsigned long long)v0 + (unsigned long long)g * sizeof(v4f);
    asm volatile("global_load_async_to_lds_b128 %0, %1, off th:TH_LOAD_NT"
                 :: "v"(lds_off), "v"(src)
                 : "memory");
  }

  for (int t = 0; t < kTilesPerBlock; ++t) {
    // ---- issue next tile's async load, then wait for the current tile ----
    if (t + 1 < kTilesPerBlock) {
      const unsigned g       = (tileBase + (unsigned)t + 1u) * (unsigned)kBlock + tid;
      const unsigned lds_off = lds_base +
          ((unsigned)((t + 1) & 1) * (unsigned)kBlock + tid) * (unsigned)sizeof(v4f);
      const unsigned long long src =
          (unsigned long long)v0 + (unsigned long long)g * sizeof(v4f);
      asm volatile("global_load_async_to_lds_b128 %0, %1, off th:TH_LOAD_NT"
                   :: "v"(lds_off), "v"(src)
                   : "memory");
      // Async loads complete in order: <=1 outstanding means current tile landed.
      asm volatile("s_wait_asynccnt 1" ::: "memory");
    } else {
      asm volatile("s_wait_asynccnt 0" ::: "memory");
    }

    const unsigned g = (tileBase + (unsigned)t) * (unsigned)kBlock + tid;

    // N=256 -> 64 float4 per (b,f) row; broadcast membrane input for this row.
    // Tiny (512 KB) and reused 256x -> regular (RT) load, stays L2-resident.
    const float s = spikes[g >> 6];

    // Current tile out of LDS (ds_load_b128), per-lane private slot.
    v4f v = buf[t & 1][tid];

    v4f vr;
#pragma unroll
    for (int i = 0; i < 4; ++i) {
      const float x = fmaf(v[i], kDecay, s);      // v = v*decay + input
      vr[i] = (x > kThresh) ? 0.0f : x;           // spike1 + reset
    }

    // Opaque register barrier: keeps the faithful post-reset compare from
    // being constant-folded (select(v>T,0,v) > T == false) by instcombine,
    // which would dead-code the entire load stream. Emits zero instructions.
#pragma unroll
    for (int i = 0; i < 4; ++i) {
      float x = vr[i];
      asm volatile("" : "+v"(x));
      vr[i] = x;
    }

    v4f o;
#pragma unroll
    for (int i = 0; i < 4; ++i) {
      o[i] = (vr[i] > kThresh) ? 1.0f : 0.0f;     // final (post-reset) spike
    }

    // Write-once output, never re-read: non-temporal 128-bit store.
    __builtin_nontemporal_store(o, reinterpret_cast<v4f*>(out) + g);
  }
}

extern "C" void kernel_launch(void* const* d_in, const int* in_sizes, int n_in,
                              void* d_out, int out_size, void* d_ws, size_t ws_size,
                              hipStream_t stream) {
  (void)in_sizes; (void)n_in; (void)d_ws; (void)ws_size; (void)out_size;
  const float* spikes = (const float*)d_in[0];   // [B, F]      f32
  const float* v0     = (const float*)d_in[1];   // [B, F, N]   f32
  float*       out    = (float*)d_out;           // [B, F, N]   f32

  lif_stream_kernel<<<dim3(kGrid), dim3(kBlock), 0, stream>>>(spikes, v0, out);
}